// TokenInitializer_36919538876844
// MI455X (gfx1250) — compile-verified
//
#include <hip/hip_runtime.h>
#include <math.h>

#define BATCH   16
#define NPTS    131072
#define NPOINT  1024
#define HID     192
#define OUTC    384
#define MROWS   (BATCH * NPOINT)      // 16384
#define OUT_ELEMS ((size_t)MROWS * OUTC)

typedef __attribute__((ext_vector_type(2))) float v2f;
typedef __attribute__((ext_vector_type(8))) float v8f;

// --------------------------------------------------------------------------
// Kernel 0: pack points into (x, y, z, min_dist=1e10) float4s in workspace.
// One b128-load/b32-store stream per point in the FPS sweep instead of
// 3x b32 point loads + separate min_dist load. Re-run every launch so the
// workspace state is deterministic across graph replays.
// --------------------------------------------------------------------------
__global__ __launch_bounds__(256) void pack_kernel(
    const float* __restrict__ points,   // [B*N, 3]
    float4* __restrict__ pts4)          // ws [B*N]
{
  const int i = blockIdx.x * 256 + threadIdx.x;   // 0 .. B*N-1
  float4 p;
  p.x = points[3 * i + 0];
  p.y = points[3 * i + 1];
  p.z = points[3 * i + 2];
  p.w = 1e10f;                                    // running min distance
  pts4[i] = p;
}

// --------------------------------------------------------------------------
// Kernel 1: persistent farthest-point-sampling. One 1024-thread block per
// batch; the packed (xyz,min_dist) array is L2-resident (32 MB of 192 MB).
// Argmax: wave32 shuffle reduction -> LDS -> wave0 reduction. Tie-break
// matches jnp.argmax (first/lowest index wins).
// --------------------------------------------------------------------------
__global__ __launch_bounds__(1024) void fps_kernel(
    float4* __restrict__ pts4,          // ws [B, N] packed
    float* __restrict__ centers)        // ws [B, NPOINT, 3]
{
  const int b    = blockIdx.x;
  const int tid  = threadIdx.x;
  const int lane = tid & 31;
  const int wid  = tid >> 5;

  float4* pb = pts4 + (size_t)b * NPTS;

  __shared__ float s_lx, s_ly, s_lz;
  __shared__ float s_rv[32];
  __shared__ int   s_ri[32];

  if (tid == 0) {
    float4 p = pb[0];
    s_lx = p.x; s_ly = p.y; s_lz = p.z;
    float* c = centers + (size_t)b * NPOINT * 3;
    c[0] = p.x; c[1] = p.y; c[2] = p.z;          // center 0 = point 0
  }
  __syncthreads();

  for (int it = 1; it < NPOINT; ++it) {
    const float lx = s_lx, ly = s_ly, lz = s_lz;
    float best = -1.0f;
    int   bidx = 0x7fffffff;

    for (int j = tid; j < NPTS; j += 1024) {
      if (j + 1024 < NPTS)
        __builtin_prefetch(&pb[j + 1024], 0, 1);  // global_prefetch_b8
      float4 p = pb[j];
      float dx = p.x - lx, dy = p.y - ly, dz = p.z - lz;
      float d  = dx * dx + dy * dy + dz * dz;
      float md = fminf(p.w, d);
      ((float*)&pb[j])[3] = md;                   // store only .w (b32)
      if (md > best) { best = md; bidx = j; }     // strict > keeps lowest j
    }

    // wave32 argmax reduction (ties -> smaller index)
    for (int off = 16; off > 0; off >>= 1) {
      float ov = __shfl_down(best, off, 32);
      int   oi = __shfl_down(bidx, off, 32);
      if (ov > best || (ov == best && oi < bidx)) { best = ov; bidx = oi; }
    }
    if (lane == 0) { s_rv[wid] = best; s_ri[wid] = bidx; }
    __syncthreads();

    if (wid == 0) {
      best = s_rv[lane];
      bidx = s_ri[lane];
      for (int off = 16; off > 0; off >>= 1) {
        float ov = __shfl_down(best, off, 32);
        int   oi = __shfl_down(bidx, off, 32);
        if (ov > best || (ov == best && oi < bidx)) { best = ov; bidx = oi; }
      }
      if (lane == 0) {
        float4 p = pb[bidx];
        s_lx = p.x; s_ly = p.y; s_lz = p.z;
        float* c = centers + ((size_t)b * NPOINT + it) * 3;
        c[0] = p.x; c[1] = p.y; c[2] = p.z;
      }
    }
    __syncthreads();
  }
}

// --------------------------------------------------------------------------
// Kernel 2: shared MLP (3 -> 192 GELU -> 384) for both heads using fp32 WMMA
// V_WMMA_F32_16X16X4_F32. One wave per (16-row tile, head). Layer-1 result
// goes through LDS to convert D-layout (rows over VGPRs) into A-layout
// (rows over lanes, K within lane) for layer 2.
// --------------------------------------------------------------------------
#define HSTRIDE 200   // 192 + 8 floats of padding against bank conflicts

__global__ __launch_bounds__(128) void mlp_wmma_kernel(
    const float* __restrict__ centers,  // [MROWS, 3]
    const float* __restrict__ w1t, const float* __restrict__ b1t,
    const float* __restrict__ w2t, const float* __restrict__ b2t,
    const float* __restrict__ w1p, const float* __restrict__ b1p,
    const float* __restrict__ w2p, const float* __restrict__ b2p,
    float* __restrict__ out)            // [2, MROWS, OUTC] concatenated
{
  __shared__ float Hs[4][16 * HSTRIDE];   // per-wave 16 x HID hidden tile

  const int tid   = threadIdx.x;
  const int lane  = tid & 31;
  const int wslot = tid >> 5;
  const int gw    = blockIdx.x * 4 + wslot;  // 0..2047
  const int mlp   = gw & 1;                  // 0 = token head, 1 = pos head
  const int mt    = gw >> 1;                 // 0..1023 (16-row tile)

  const float* w1 = mlp ? w1p : w1t;
  const float* b1 = mlp ? b1p : b1t;
  const float* w2 = mlp ? w2p : w2t;
  const float* b2 = mlp ? b2p : b2t;
  float* o = out + (size_t)mlp * OUT_ELEMS;

  const int half = lane >> 4;    // 0: K={0,1}   1: K={2,3}
  const int l16  = lane & 15;
  const int k0   = half * 2;

  // ---- A fragment for layer 1: 16x4 tile of centers (K=3, padded to 4) ----
  const int row = mt * 16 + l16;
  const float cx = centers[row * 3 + 0];
  const float cy = centers[row * 3 + 1];
  const float cz = centers[row * 3 + 2];
  v2f a1;
  a1.x = half ? cz : cx;          // K = 0 or 2
  a1.y = half ? 0.0f : cy;        // K = 1 or 3 (pad)

  // ---- Layer 1: H = GELU(centers @ w1 + b1), staged into LDS ----
  for (int nt = 0; nt < HID / 16; ++nt) {
    const int col = nt * 16 + l16;
    // load both needed w1 rows unconditionally (rows 0..2 all exist),
    // then select per-lane: avoids exec-masked loads around the WMMA.
    const float w1_k0 = w1[k0 * HID + col];        // row 0 or row 2
    const float w1_r1 = w1[1 * HID + col];         // row 1 (always valid)
    v2f bf;
    bf.x = w1_k0;
    bf.y = half ? 0.0f : w1_r1;                    // K=1 valid, K=3 pad
    v8f c = {};
    c = __builtin_amdgcn_wmma_f32_16x16x4_f32(
        /*neg_a=*/false, a1, /*neg_b=*/false, bf,
        /*c_mod=*/(short)0, c, /*reuse_a=*/false, /*reuse_b=*/false);
    const float bias = b1[col];
#pragma unroll
    for (int v = 0; v < 8; ++v) {
      float x = c[v] + bias;
      // exact GELU: x * 0.5 * (1 + erf(x / sqrt(2)))
      float g = 0.5f * x * (1.0f + erff(x * 0.70710678118654752f));
      const int r = v + half * 8;                  // D-layout row
      Hs[wslot][r * HSTRIDE + col] = g;
    }
  }
  // same-wave DS store->load ordering is enforced via DScnt by the compiler

  // ---- Layer 2: out = H @ w2 + b2, K = 192 in 48 WMMA steps ----
  const float* Hrow = &Hs[wslot][l16 * HSTRIDE];   // A-layout row for lane
  for (int nt = 0; nt < OUTC / 16; ++nt) {
    const int col = nt * 16 + l16;
    v8f acc = {};
#pragma unroll 4
    for (int ks = 0; ks < HID / 4; ++ks) {
      const int kk = ks * 4 + k0;
      v2f af; af.x = Hrow[kk];            af.y = Hrow[kk + 1];
      v2f bf; bf.x = w2[kk * OUTC + col]; bf.y = w2[(kk + 1) * OUTC + col];
      acc = __builtin_amdgcn_wmma_f32_16x16x4_f32(
          false, af, false, bf, (short)0, acc, false, false);
    }
    const float bias = b2[col];
#pragma unroll
    for (int v = 0; v < 8; ++v) {
      const int r = mt * 16 + v + half * 8;        // D-layout row
      o[(size_t)r * OUTC + col] = acc[v] + bias;
    }
  }
}

// --------------------------------------------------------------------------
extern "C" void kernel_launch(void* const* d_in, const int* in_sizes, int n_in,
                              void* d_out, int out_size, void* d_ws, size_t ws_size,
                              hipStream_t stream) {
  (void)in_sizes; (void)n_in; (void)out_size; (void)ws_size;

  const float* points = (const float*)d_in[0];
  const float* w1t = (const float*)d_in[1];
  const float* b1t = (const float*)d_in[2];
  const float* w2t = (const float*)d_in[3];
  const float* b2t = (const float*)d_in[4];
  const float* w1p = (const float*)d_in[5];
  const float* b1p = (const float*)d_in[6];
  const float* w2p = (const float*)d_in[7];
  const float* b2p = (const float*)d_in[8];

  // workspace layout: [B*N] float4 packed (xyz, min_dist) | [B*NPOINT*3] centers
  float4* pts4   = (float4*)d_ws;
  float* centers = (float*)(pts4 + (size_t)BATCH * NPTS);

  pack_kernel<<<(BATCH * NPTS) / 256, 256, 0, stream>>>(points, pts4);
  fps_kernel<<<BATCH, 1024, 0, stream>>>(pts4, centers);

  // 2048 waves = 1024 M-tiles x 2 heads; 4 waves / 128-thread block
  mlp_wmma_kernel<<<512, 128, 0, stream>>>(
      centers, w1t, b1t, w2t, b2t, w1p, b1p, w2p, b2p, (float*)d_out);
}